// SimpleConv_70789650972922
// MI455X (gfx1250) — compile-verified
//
#include <hip/hip_runtime.h>

// ---------------------------------------------------------------------------
// SimpleConv for MI455X (gfx1250, wave32):
//   h   = feat @ W            -> WMMA f32 16x16x4 GEMM (h in d_ws)
//   agg = segment_sum(h[src]*ew, dst)  -> f32 global atomics
//   out = relu(agg)
// Memory-bound workload (~1 GB traffic -> ~45us floor @ 23.3 TB/s), so we keep
// full f32 precision and use the f32-native CDNA5 WMMA for the projection.
// ---------------------------------------------------------------------------

typedef __attribute__((ext_vector_type(2))) float v2f;
typedef __attribute__((ext_vector_type(8))) float v8f;

#define N_NODES   100000
#define N_EDGES   1600000
#define IN_FEATS  256
#define OUT_FEATS 64

// LDS holds W transposed: Wt[n][k], padded row stride 258 floats so that
//  - staging writes (fixed k, n varies across lanes) hit distinct banks
//  - per-fragment ds_load_b64 at Wt[n*258 + kb] is conflict-free across lanes
#define WT_STRIDE 258   // 64 rows * 258 * 4B = 66,048 bytes dynamic LDS

// ------------------------------ GEMM --------------------------------------
// One wave (32 lanes) computes a 16(M) x 64(N) strip of h = feat @ W.
// 100000 rows = 6250 exact 16-row tiles. Block = 256 threads = 8 waves.
__global__ __launch_bounds__(256)
void gemm_wmma_f32(const float* __restrict__ feat,
                   const float* __restrict__ W,
                   float* __restrict__ h)
{
    extern __shared__ float Wt[];   // [64][WT_STRIDE]
    const int tid = threadIdx.x;

    // Stage W (256x64) transposed into LDS: coalesced global b32 reads.
    for (int i = tid; i < IN_FEATS * OUT_FEATS; i += 256) {
        const int k = i >> 6;       // 0..255
        const int n = i & 63;       // 0..63
        Wt[n * WT_STRIDE + k] = W[i];
    }
    __syncthreads();

    const int lane  = tid & 31;
    const int wave  = tid >> 5;
    const int mtile = blockIdx.x * 8 + wave;
    if (mtile >= N_NODES / 16) return;   // wave-uniform: EXEC stays all-ones

    const int m0   = mtile * 16;
    const int half = lane >> 4;          // 0: lanes 0-15, 1: lanes 16-31
    const int l16  = lane & 15;

    // A-fragment layout (ISA 32-bit A 16x4): lane L holds row M=L%16,
    // VGPR0/1 = K = {0,1} for lanes 0-15, {2,3} for lanes 16-31.
    const float* arow = feat + (size_t)(m0 + l16) * IN_FEATS + 2 * half;

    v8f acc0 = {}, acc1 = {}, acc2 = {}, acc3 = {};

    for (int k0 = 0; k0 < IN_FEATS; k0 += 4) {
        const v2f a = *(const v2f*)(arow + k0);          // global_load_b64
        const int kb = k0 + 2 * half;

        // B-fragment (4x16, KxN): lanes 0-15 carry K={k0,k0+1}, lanes 16-31
        // carry K={k0+2,k0+3}; N = n0 + lane%16. Transposed LDS makes each
        // fragment one ds_load_b64.
        const v2f b0 = *(const v2f*)(Wt + (0  + l16) * WT_STRIDE + kb);
        const v2f b1 = *(const v2f*)(Wt + (16 + l16) * WT_STRIDE + kb);
        const v2f b2 = *(const v2f*)(Wt + (32 + l16) * WT_STRIDE + kb);
        const v2f b3 = *(const v2f*)(Wt + (48 + l16) * WT_STRIDE + kb);

        acc0 = __builtin_amdgcn_wmma_f32_16x16x4_f32(false, a, false, b0,
                                                     (short)0, acc0, false, false);
        acc1 = __builtin_amdgcn_wmma_f32_16x16x4_f32(false, a, false, b1,
                                                     (short)0, acc1, false, false);
        acc2 = __builtin_amdgcn_wmma_f32_16x16x4_f32(false, a, false, b2,
                                                     (short)0, acc2, false, false);
        acc3 = __builtin_amdgcn_wmma_f32_16x16x4_f32(false, a, false, b3,
                                                     (short)0, acc3, false, false);
    }

    // D layout (ISA 32-bit C/D 16x16): VGPR v holds row m0 + v + 8*half,
    // col = n0 + lane%16.
    #pragma unroll
    for (int v = 0; v < 8; ++v) {
        const size_t row = (size_t)(m0 + v + 8 * half);
        float* const o = h + row * OUT_FEATS + l16;
        o[0]  = acc0[v];
        o[16] = acc1[v];
        o[32] = acc2[v];
        o[48] = acc3[v];
    }
}

// --------------------------- zero / scatter / relu ------------------------
__global__ __launch_bounds__(256)
void zero_f4(float4* __restrict__ out, int n4)
{
    const int i = blockIdx.x * 256 + threadIdx.x;
    if (i < n4) out[i] = make_float4(0.f, 0.f, 0.f, 0.f);
}

// One thread per (edge, out-column). 64 consecutive threads share one edge,
// so each wave32 has a uniform edge index (scalar-friendly) and the h-gather
// and atomic stores are coalesced across the 64-float output row.
__global__ __launch_bounds__(256)
void edge_scatter(const float* __restrict__ h,
                  const float* __restrict__ ew,
                  const int*   __restrict__ src,
                  const int*   __restrict__ dst,
                  float* __restrict__ out)
{
    const int tid = blockIdx.x * 256 + threadIdx.x;   // < E*64 = 102.4M < 2^31
    const int e = tid >> 6;
    const int c = tid & 63;
    if (e >= N_EDGES) return;

    const int   s = src[e];
    const int   d = dst[e];
    const float w = ew[e];
    const float val = h[(size_t)s * OUT_FEATS + c] * w;
    atomicAdd(out + (size_t)d * OUT_FEATS + c, val);
}

__global__ __launch_bounds__(256)
void relu_f4(float4* __restrict__ out, int n4)
{
    const int i = blockIdx.x * 256 + threadIdx.x;
    if (i < n4) {
        float4 v = out[i];
        v.x = v.x > 0.f ? v.x : 0.f;
        v.y = v.y > 0.f ? v.y : 0.f;
        v.z = v.z > 0.f ? v.z : 0.f;
        v.w = v.w > 0.f ? v.w : 0.f;
        out[i] = v;
    }
}

// ------------------------------ launch ------------------------------------
extern "C" void kernel_launch(void* const* d_in, const int* in_sizes, int n_in,
                              void* d_out, int out_size, void* d_ws, size_t ws_size,
                              hipStream_t stream)
{
    const float* feat = (const float*)d_in[0];   // [100000, 256]
    const float* W    = (const float*)d_in[1];   // [256, 64]
    const float* ew   = (const float*)d_in[2];   // [1600000]
    const int*   src  = (const int*)  d_in[3];   // [1600000]
    const int*   dst  = (const int*)  d_in[4];   // [1600000]
    float*       out  = (float*)d_out;           // [100000, 64]
    float*       h    = (float*)d_ws;            // scratch: 25.6 MB

    const int out_f4 = (N_NODES * OUT_FEATS) / 4;          // 1.6M float4
    zero_f4<<<(out_f4 + 255) / 256, 256, 0, stream>>>((float4*)out, out_f4);

    const int mtiles  = N_NODES / 16;                      // 6250 (exact)
    const int gblocks = (mtiles + 7) / 8;                  // 8 waves/block
    gemm_wmma_f32<<<gblocks, 256, OUT_FEATS * WT_STRIDE * sizeof(float), stream>>>(
        feat, W, h);

    const long long sthreads = (long long)N_EDGES * OUT_FEATS;   // 102.4M
    const int sblocks = (int)((sthreads + 255) / 256);           // 400000
    edge_scatter<<<sblocks, 256, 0, stream>>>(h, ew, src, dst, out);

    relu_f4<<<(out_f4 + 255) / 256, 256, 0, stream>>>((float4*)out, out_f4);
}